// DR2FWL2Conv_2302102471410
// MI455X (gfx1250) — compile-verified
//
#include <hip/hip_runtime.h>

typedef float v2f __attribute__((ext_vector_type(2)));
typedef float v4f __attribute__((ext_vector_type(4)));
typedef float v8f __attribute__((ext_vector_type(8)));

#define AT_STRIDE 68   // 16x64 tile padded to 68 floats/row: 16B aligned, bank-conflict-free

// ---------------------------------------------------------------------------
// zero workspace (float4 granularity)
// ---------------------------------------------------------------------------
__global__ void zero_kernel(float* __restrict__ p, long long n4) {
    v4f z = {0.f, 0.f, 0.f, 0.f};
    long long stride = (long long)gridDim.x * blockDim.x;
    for (long long i = (long long)blockIdx.x * blockDim.x + threadIdx.x; i < n4; i += stride)
        ((v4f*)p)[i] = z;
}

// ---------------------------------------------------------------------------
// Generic triangle branch:  acc[idxOut[t]] += relu((srcA[idxA[t]]+srcB[idxB[t]]) @ W + bias)
// One wave processes a tile of 16 triangles using V_WMMA_F32_16X16X4_F32.
// ---------------------------------------------------------------------------
__global__ __launch_bounds__(256) void tri_kernel(
    const float* __restrict__ srcA, const int* __restrict__ idxA,
    const float* __restrict__ srcB, const int* __restrict__ idxB,
    const float* __restrict__ W,    const float* __restrict__ bias,
    const int* __restrict__ idxOut, float* __restrict__ acc, int T) {

    __shared__ __align__(16) float lds[64 * 64 + 64 + 8 * 16 * AT_STRIDE];
    float* Wl = lds;             // 64x64, row-major [k][n]
    float* Bl = lds + 4096;      // 64
    float* Ap = lds + 4160;      // 8 waves * 16x68

    const int tid = threadIdx.x;
    for (int i = tid; i < 1024; i += 256) ((v4f*)Wl)[i] = ((const v4f*)W)[i];
    if (tid < 16) ((v4f*)Bl)[tid] = ((const v4f*)bias)[tid];
    __syncthreads();

    const int wave = tid >> 5, lane = tid & 31;
    const int l16 = lane & 15, half = lane >> 4;
    float* At = Ap + wave * (16 * AT_STRIDE);

    const int nTiles = (T + 15) >> 4;
    const int nw = gridDim.x * 8;
    for (int tile = blockIdx.x * 8 + wave; tile < nTiles; tile += nw) {
        const int t16 = tile << 4;
        int rr = t16 + l16; if (rr >= T) rr = T - 1;
        const int iA = idxA[rr], iB = idxB[rr], iO = idxOut[rr];

        // gather + sum two rows -> LDS tile (coalesced b128 per lane)
        #pragma unroll
        for (int j = 0; j < 8; ++j) {
            const int row = j * 2 + half;
            const long long ra = __shfl(iA, row);
            const long long rb = __shfl(iB, row);
            v4f a = *(const v4f*)(srcA + ra * 64 + l16 * 4);
            v4f b = *(const v4f*)(srcB + rb * 64 + l16 * 4);
            *(v4f*)(At + row * AT_STRIDE + l16 * 4) = a + b;
        }
        asm volatile("s_wait_dscnt 0" ::: "memory");

        // A fragments: 16x4 f32 layout (lanes 0-15: K=4s,4s+1; lanes 16-31: K=4s+2,4s+3)
        v2f af[16];
        #pragma unroll
        for (int s = 0; s < 16; ++s)
            af[s] = *(const v2f*)(At + l16 * AT_STRIDE + s * 4 + half * 2);

        #pragma unroll
        for (int n0 = 0; n0 < 64; n0 += 16) {
            v8f c = {0.f, 0.f, 0.f, 0.f, 0.f, 0.f, 0.f, 0.f};
            #pragma unroll
            for (int s = 0; s < 16; ++s) {
                v2f bf;
                bf.x = Wl[(4 * s + 2 * half + 0) * 64 + n0 + l16];
                bf.y = Wl[(4 * s + 2 * half + 1) * 64 + n0 + l16];
                c = __builtin_amdgcn_wmma_f32_16x16x4_f32(false, af[s], false, bf,
                                                          (short)0, c, false, false);
            }
            const float bn = Bl[n0 + l16];
            #pragma unroll
            for (int r = 0; r < 8; ++r) {
                const int m = r + 8 * half;                 // row inside the tile
                float v = c[r] + bn;
                v = v > 0.f ? v : 0.f;
                const int orow = __shfl(iO, m);
                if (t16 + m < T)
                    atomicAdd(acc + (long long)orow * 64 + n0 + l16, v);
            }
        }
    }
}

// ---------------------------------------------------------------------------
// acc[i] += a[i] + a[inv[i]]   (rows of 64 floats, float4 vectorized)
// ---------------------------------------------------------------------------
__global__ void combine_kernel(float* __restrict__ acc, const float* __restrict__ a,
                               const int* __restrict__ inv, int N) {
    const long long n4 = (long long)N * 16;
    v4f* acc4 = (v4f*)acc;
    const v4f* a4 = (const v4f*)a;
    const long long stride = (long long)gridDim.x * blockDim.x;
    for (long long i = (long long)blockIdx.x * blockDim.x + threadIdx.x; i < n4; i += stride) {
        const long long row = i >> 4;
        const int c = (int)(i & 15);
        const long long irow = inv[row];
        acc4[i] = acc4[i] + a4[i] + a4[irow * 16 + c];
    }
}

// ---------------------------------------------------------------------------
// MLP pass A:  z = relu(((1+eps)*edge + acc) @ W1 + b1) @ W2 + b2  -> out
// and per-channel sum / sum-of-squares -> stats (64*2 floats)
// 128 threads = 4 waves; wave-per-16-row tile, WMMA f32 16x16x4.
// ---------------------------------------------------------------------------
__global__ __launch_bounds__(128) void mlp_kernel(
    const float* __restrict__ edge, const float* __restrict__ acc,
    const float* __restrict__ epsPtr,
    const float* __restrict__ W1, const float* __restrict__ b1,
    const float* __restrict__ W2, const float* __restrict__ b2,
    float* __restrict__ out, float* __restrict__ stats, int N) {

    __shared__ __align__(16) float lds[4096 * 2 + 64 * 2 + 4 * 16 * AT_STRIDE];
    float* W1l = lds;
    float* W2l = lds + 4096;
    float* b1l = lds + 8192;
    float* b2l = lds + 8256;
    float* Ap  = lds + 8320;

    const int tid = threadIdx.x;
    for (int i = tid; i < 1024; i += 128) {
        ((v4f*)W1l)[i] = ((const v4f*)W1)[i];
        ((v4f*)W2l)[i] = ((const v4f*)W2)[i];
    }
    if (tid < 16)       ((v4f*)b1l)[tid]      = ((const v4f*)b1)[tid];
    else if (tid < 32)  ((v4f*)b2l)[tid - 16] = ((const v4f*)b2)[tid - 16];
    __syncthreads();

    const float epsv = 1.0f + epsPtr[0];
    const int wave = tid >> 5, lane = tid & 31;
    const int l16 = lane & 15, half = lane >> 4;
    float* At = Ap + wave * (16 * AT_STRIDE);

    float s1[4] = {0.f, 0.f, 0.f, 0.f};
    float s2[4] = {0.f, 0.f, 0.f, 0.f};

    const int nTiles = (N + 15) >> 4;
    const int nw = gridDim.x * 4;
    for (int tile = blockIdx.x * 4 + wave; tile < nTiles; tile += nw) {
        const int t16 = tile << 4;

        // y = (1+eps)*edge + acc  -> LDS tile
        #pragma unroll
        for (int j = 0; j < 8; ++j) {
            const int row = j * 2 + half;
            long long gr = t16 + row; if (gr >= N) gr = N - 1;
            v4f e = *(const v4f*)(edge + gr * 64 + l16 * 4);
            v4f a = *(const v4f*)(acc  + gr * 64 + l16 * 4);
            *(v4f*)(At + row * AT_STRIDE + l16 * 4) = e * epsv + a;
        }
        asm volatile("s_wait_dscnt 0" ::: "memory");

        v2f af[16];
        #pragma unroll
        for (int s = 0; s < 16; ++s)
            af[s] = *(const v2f*)(At + l16 * AT_STRIDE + s * 4 + half * 2);

        // GEMM1 + bias + relu; write h back into the (now free) A tile
        #pragma unroll
        for (int n0 = 0; n0 < 64; n0 += 16) {
            v8f c = {0.f, 0.f, 0.f, 0.f, 0.f, 0.f, 0.f, 0.f};
            #pragma unroll
            for (int s = 0; s < 16; ++s) {
                v2f bf;
                bf.x = W1l[(4 * s + 2 * half + 0) * 64 + n0 + l16];
                bf.y = W1l[(4 * s + 2 * half + 1) * 64 + n0 + l16];
                c = __builtin_amdgcn_wmma_f32_16x16x4_f32(false, af[s], false, bf,
                                                          (short)0, c, false, false);
            }
            const float bn = b1l[n0 + l16];
            #pragma unroll
            for (int r = 0; r < 8; ++r) {
                const int m = r + 8 * half;
                float h = c[r] + bn;
                h = h > 0.f ? h : 0.f;
                At[m * AT_STRIDE + n0 + l16] = h;   // layout transpose via LDS
            }
        }
        asm volatile("s_wait_dscnt 0" ::: "memory");

        v2f hf[16];
        #pragma unroll
        for (int s = 0; s < 16; ++s)
            hf[s] = *(const v2f*)(At + l16 * AT_STRIDE + s * 4 + half * 2);

        // GEMM2 + bias; write z to out, accumulate channel stats
        #pragma unroll
        for (int n0 = 0; n0 < 64; n0 += 16) {
            const int q = n0 >> 4;
            v8f c = {0.f, 0.f, 0.f, 0.f, 0.f, 0.f, 0.f, 0.f};
            #pragma unroll
            for (int s = 0; s < 16; ++s) {
                v2f bf;
                bf.x = W2l[(4 * s + 2 * half + 0) * 64 + n0 + l16];
                bf.y = W2l[(4 * s + 2 * half + 1) * 64 + n0 + l16];
                c = __builtin_amdgcn_wmma_f32_16x16x4_f32(false, hf[s], false, bf,
                                                          (short)0, c, false, false);
            }
            const float bn = b2l[n0 + l16];
            #pragma unroll
            for (int r = 0; r < 8; ++r) {
                const int m = r + 8 * half;
                const long long gr = t16 + m;
                if (gr < N) {
                    const float z = c[r] + bn;
                    out[gr * 64 + n0 + l16] = z;
                    s1[q] += z;
                    s2[q] += z * z;
                }
            }
        }
    }

    // fold the two half-wave copies of each channel, then 8 atomics per wave
    #pragma unroll
    for (int q = 0; q < 4; ++q) {
        float a1 = s1[q] + __shfl_xor(s1[q], 16);
        float a2 = s2[q] + __shfl_xor(s2[q], 16);
        if (half == 0) {
            atomicAdd(stats + (q * 16 + l16) * 2 + 0, a1);
            atomicAdd(stats + (q * 16 + l16) * 2 + 1, a2);
        }
    }
}

// ---------------------------------------------------------------------------
// BN finalize (in place on out): out = g*(z-mean)*rsqrt(var+eps)+b
// ---------------------------------------------------------------------------
__global__ void bn_kernel(float* __restrict__ out, const float* __restrict__ stats,
                          const float* __restrict__ g, const float* __restrict__ b, int N) {
    __shared__ float scale[64], shift[64];
    const int tid = threadIdx.x;
    if (tid < 64) {
        const float invN = 1.0f / (float)N;
        const float m = stats[tid * 2] * invN;
        float var = stats[tid * 2 + 1] * invN - m * m;
        var = var > 0.f ? var : 0.f;
        const float sc = g[tid] * rsqrtf(var + 1e-5f);
        scale[tid] = sc;
        shift[tid] = b[tid] - m * sc;
    }
    __syncthreads();

    const long long n4 = (long long)N * 16;
    v4f* o4 = (v4f*)out;
    const long long stride = (long long)gridDim.x * blockDim.x;
    for (long long i = (long long)blockIdx.x * blockDim.x + threadIdx.x; i < n4; i += stride) {
        const int c4 = (int)(i & 15) * 4;
        v4f z = o4[i], r;
        r.x = z.x * scale[c4 + 0] + shift[c4 + 0];
        r.y = z.y * scale[c4 + 1] + shift[c4 + 1];
        r.z = z.z * scale[c4 + 2] + shift[c4 + 2];
        r.w = z.w * scale[c4 + 3] + shift[c4 + 3];
        o4[i] = r;
    }
}

// ---------------------------------------------------------------------------
// launch
// ---------------------------------------------------------------------------
extern "C" void kernel_launch(void* const* d_in, const int* in_sizes, int n_in,
                              void* d_out, int out_size, void* d_ws, size_t ws_size,
                              hipStream_t stream) {
    (void)n_in; (void)out_size; (void)ws_size;

    const float* e0 = (const float*)d_in[0];
    const float* e1 = (const float*)d_in[1];
    const float* e2 = (const float*)d_in[2];
    const int* tri011 = (const int*)d_in[6];
    const int* tri111 = (const int*)d_in[7];
    const int* tri112 = (const int*)d_in[8];
    const int* tri122 = (const int*)d_in[9];
    const int* tri222 = (const int*)d_in[10];
    const int* inv1 = (const int*)d_in[11];
    const int* inv2 = (const int*)d_in[12];
    auto P = [&](int i) { return (const float*)d_in[i]; };
    // params base 13, 9 per group:
    // +0 projW, +1 projb, +2 mlpW1, +3 mlpb1, +4 mlpW2, +5 mlpb2, +6 normg, +7 normb, +8 eps

    const int N0 = in_sizes[0] / 64, N1 = in_sizes[1] / 64, N2 = in_sizes[2] / 64;
    const int T = in_sizes[6] / 3;

    float* ws   = (float*)d_ws;
    float* acc0 = ws;
    float* acc1 = acc0 + (long long)N0 * 64;
    float* acc2 = acc1 + (long long)N1 * 64;
    float* a112 = acc2 + (long long)N2 * 64;
    float* a212 = a112 + (long long)N1 * 64;
    float* stats = a212 + (long long)N2 * 64;   // 3 * 128 floats

    const long long totWs = (long long)(N0 + 2LL * N1 + 2LL * N2) * 64 + 384;
    zero_kernel<<<2048, 256, 0, stream>>>(ws, (totWs + 3) / 4);

    const int triTiles = (T + 15) >> 4;
    int triGrid = (triTiles + 7) / 8; if (triGrid > 4096) triGrid = 4096;

    // x011: (e1[ik]+e1[ik]) @ projW0 -> acc0 @ ij
    tri_kernel<<<triGrid, 256, 0, stream>>>(e1, tri011 + T, e1, tri011 + T, P(13), P(14), tri011, acc0, T);
    // x111: (e1[ik]+e1[kj]) @ projW1 -> acc1 @ ij
    tri_kernel<<<triGrid, 256, 0, stream>>>(e1, tri111 + T, e1, tri111 + 2 * T, P(22), P(23), tri111, acc1, T);
    // x112: (e1[ik]+e2[kj]) @ projW1 -> a112 @ ij
    tri_kernel<<<triGrid, 256, 0, stream>>>(e1, tri112 + T, e2, tri112 + 2 * T, P(22), P(23), tri112, a112, T);
    // x211: (e1[ij]+e1[ik]) @ projW2 -> acc2 @ kj
    tri_kernel<<<triGrid, 256, 0, stream>>>(e1, tri112, e1, tri112 + T, P(31), P(32), tri112 + 2 * T, acc2, T);
    // x122: (e2[ik]+e2[kj]) @ projW1 -> acc1 @ ij
    tri_kernel<<<triGrid, 256, 0, stream>>>(e2, tri122 + T, e2, tri122 + 2 * T, P(22), P(23), tri122, acc1, T);
    // x212: (e1[ij]+e2[kj]) @ projW2 -> a212 @ ik
    tri_kernel<<<triGrid, 256, 0, stream>>>(e1, tri122, e2, tri122 + 2 * T, P(31), P(32), tri122 + T, a212, T);
    // x222: (e2[ik]+e2[kj]) @ projW2 -> acc2 @ ij
    tri_kernel<<<triGrid, 256, 0, stream>>>(e2, tri222 + T, e2, tri222 + 2 * T, P(31), P(32), tri222, acc2, T);

    combine_kernel<<<2048, 256, 0, stream>>>(acc1, a112, inv1, N1);
    combine_kernel<<<2048, 256, 0, stream>>>(acc2, a212, inv2, N2);

    float* out0 = (float*)d_out;
    float* out1 = out0 + (long long)N0 * 64;
    float* out2 = out1 + (long long)N1 * 64;

    auto mlpGrid = [](int N) { int g = (((N + 15) >> 4) + 3) / 4; return g > 4096 ? 4096 : g; };
    mlp_kernel<<<mlpGrid(N0), 128, 0, stream>>>(e0, acc0, P(21), P(15), P(16), P(17), P(18), out0, stats,       N0);
    mlp_kernel<<<mlpGrid(N1), 128, 0, stream>>>(e1, acc1, P(30), P(24), P(25), P(26), P(27), out1, stats + 128, N1);
    mlp_kernel<<<mlpGrid(N2), 128, 0, stream>>>(e2, acc2, P(39), P(33), P(34), P(35), P(36), out2, stats + 256, N2);

    bn_kernel<<<2048, 256, 0, stream>>>(out0, stats,       P(19), P(20), N0);
    bn_kernel<<<2048, 256, 0, stream>>>(out1, stats + 128, P(28), P(29), N1);
    bn_kernel<<<2048, 256, 0, stream>>>(out2, stats + 256, P(37), P(38), N2);
}